// Loss_46883863003176
// MI455X (gfx1250) — compile-verified
//
#include <hip/hip_runtime.h>
#include <hip/hip_bf16.h>

typedef __attribute__((ext_vector_type(2))) float v2f;
typedef __attribute__((ext_vector_type(8))) float v8f;

// ---------------------------------------------------------------------------
// Wave32 all-reduce sum using V_WMMA_F32_16X16X4_F32 with an all-ones B.
// Pass 1: D[m][n] = v[m] + v[m+16]  (A vgpr0 = per-lane value, vgpr1 = 0)
//   per-lane sum of the 8 D vgprs -> T1 in lanes 0-15, T2 in lanes 16-31,
//   with T1 + T2 == total.
// Pass 2: same construction on {T1,T2} -> D[m][n] = T1 + T2 = total, in every
//   lane. Full f32 precision, EXEC must be all-ones (callers guarantee no
//   divergence at the call site).
// ---------------------------------------------------------------------------
__device__ __forceinline__ float wave32_allreduce_sum(float v) {
#if __has_builtin(__builtin_amdgcn_wmma_f32_16x16x4_f32)
    v2f a;  a[0] = v;    a[1] = 0.0f;
    v2f b;  b[0] = 1.0f; b[1] = 1.0f;
    v8f c = {};
    v8f d1 = __builtin_amdgcn_wmma_f32_16x16x4_f32(
        /*neg_a=*/false, a, /*neg_b=*/false, b,
        /*c_mod=*/(short)0, c, /*reuse_a=*/false, /*reuse_b=*/false);
    float r = ((d1[0] + d1[1]) + (d1[2] + d1[3])) +
              ((d1[4] + d1[5]) + (d1[6] + d1[7]));
    v2f a2; a2[0] = r; a2[1] = 0.0f;
    v8f d2 = __builtin_amdgcn_wmma_f32_16x16x4_f32(
        false, a2, false, b, (short)0, c, false, false);
    return d2[0];
#else
    for (int off = 16; off > 0; off >>= 1)
        v += __shfl_xor(v, off, 32);
    return v;
#endif
}

// Block-level reduction: wave WMMA all-reduce, then 8 wave partials via LDS.
__device__ __forceinline__ float block_reduce_sum(float v) {
    __shared__ float wsum[8];                 // 256 threads = 8 wave32s
    const int lane = threadIdx.x & 31;
    const int wid  = threadIdx.x >> 5;
    float t = wave32_allreduce_sum(v);
    if (lane == 0) wsum[wid] = t;
    __syncthreads();
    float s = 0.0f;
    if (threadIdx.x == 0) {
        const int nw = (blockDim.x + 31) >> 5;
        for (int i = 0; i < nw; ++i) s += wsum[i];
    }
    return s;                                  // valid in thread 0 only
}

// ---------------------------------------------------------------------------
// Stage 1: grid-stride squared-difference reduction with b128 loads.
// ~134 MB streamed; HBM-bound (~5.8 us at 23.3 TB/s).
// ---------------------------------------------------------------------------
__global__ __launch_bounds__(256)
void sqdiff_partial_kernel(const float* __restrict__ pred,
                           const float* __restrict__ targ,
                           unsigned long long n,           // total floats
                           float* __restrict__ partials) {
    const unsigned long long n4     = n >> 2;
    const unsigned long long tid    = (unsigned long long)blockIdx.x * blockDim.x + threadIdx.x;
    const unsigned long long stride = (unsigned long long)gridDim.x * blockDim.x;

    const float4* __restrict__ p4 = (const float4*)pred;
    const float4* __restrict__ t4 = (const float4*)targ;

    float acc = 0.0f;
    for (unsigned long long i = tid; i < n4; i += stride) {
        // Prefetch the next grid-stride chunk (global_prefetch_b8).
        if (i + stride < n4) {
            __builtin_prefetch(&p4[i + stride], 0, 0);
            __builtin_prefetch(&t4[i + stride], 0, 0);
        }
        float4 a = p4[i];
        float4 b = t4[i];
        float dx = a.x - b.x;
        float dy = a.y - b.y;
        float dz = a.z - b.z;
        float dw = a.w - b.w;
        acc = fmaf(dx, dx, acc);
        acc = fmaf(dy, dy, acc);
        acc = fmaf(dz, dz, acc);
        acc = fmaf(dw, dw, acc);
    }
    // Scalar tail (n not divisible by 4 — not hit for this shape, but safe).
    for (unsigned long long j = (n4 << 2) + tid; j < n; j += stride) {
        float d = pred[j] - targ[j];
        acc = fmaf(d, d, acc);
    }

    float bsum = block_reduce_sum(acc);
    if (threadIdx.x == 0) partials[blockIdx.x] = bsum;
}

// ---------------------------------------------------------------------------
// Stage 2: single-block reduction of per-block partials; scale by 1/(S*B).
// d and batch_size arrive as device-side single-element int arrays.
// ---------------------------------------------------------------------------
__global__ __launch_bounds__(256)
void final_reduce_kernel(const float* __restrict__ partials, int nPartials,
                         const int* __restrict__ dptr,
                         const int* __restrict__ bptr,
                         float* __restrict__ out) {
    float acc = 0.0f;
    for (int i = threadIdx.x; i < nPartials; i += blockDim.x)
        acc += partials[i];

    float total = block_reduce_sum(acc);
    if (threadIdx.x == 0) {
        const float S = (float)(2 * dptr[0] + 1);
        const float B = (float)bptr[0];
        out[0] = total / S / B;
    }
}

extern "C" void kernel_launch(void* const* d_in, const int* in_sizes, int n_in,
                              void* d_out, int out_size, void* d_ws, size_t ws_size,
                              hipStream_t stream) {
    const float* pred = (const float*)d_in[0];
    const float* targ = (const float*)d_in[1];
    const int*   dptr = (const int*)d_in[2];
    const int*   bptr = (const int*)d_in[3];

    const unsigned long long n = (unsigned long long)in_sizes[0];

    // 2048 blocks x 256 threads = 524288 threads; 8 float4 pairs each.
    int blocks = 2048;
    const size_t maxPartials = ws_size / sizeof(float);
    if (maxPartials == 0) blocks = 1;                 // degenerate guard
    else if ((size_t)blocks > maxPartials) blocks = (int)maxPartials;

    float* partials = (float*)d_ws;

    sqdiff_partial_kernel<<<blocks, 256, 0, stream>>>(pred, targ, n, partials);
    final_reduce_kernel<<<1, 256, 0, stream>>>(partials, blocks, dptr, bptr,
                                               (float*)d_out);
}